// FsodFastRCNNOutputLayers_50474455662706
// MI455X (gfx1250) — compile-verified
//
#include <hip/hip_runtime.h>
#include <hip/hip_bf16.h>
#include <math.h>

#define CLS_NUM   20
#define BOX_NUM   400
#define NBOX      (CLS_NUM * BOX_NUM)   // 8000
#define NPAD      8192                  // next pow2 for bitonic sort
#define COLW      (NBOX / 32)           // 250 mask words per row
#define ROWT      (NBOX / 16)           // 500 row tiles
#define COLT      (NBOX / 64)           // 125 col tiles (64 cols each)
#define TOPK      100
#define IMG_Wf    1333.0f
#define IMG_Hf    800.0f
#define SCORE_TH  0.05f
#define NMS_TH    0.5f
#define CLS_OFF   1334.0f               // > max clipped coord, per-class box offset
#define NEG_BIG   -1e30f

typedef __attribute__((ext_vector_type(2))) float v2f;
typedef __attribute__((ext_vector_type(8))) float v8f;

__device__ __forceinline__ unsigned ballot32(bool p) {
#if __has_builtin(__builtin_amdgcn_ballot_w32)
  return (unsigned)__builtin_amdgcn_ballot_w32(p);   // v_cmp -> SGPR, wave32
#else
  return (unsigned)__ballot(p);
#endif
}

// ---------------------------------------------------------------------------
// K1: transpose-interleave, finite check, clip, score mask -> sort keys
// t = box*CLS_NUM + cls  maps to original o = cls*BOX_NUM + box
// ---------------------------------------------------------------------------
__global__ void prep_kernel(const int* __restrict__ pred_cls,
                            const float* __restrict__ boxes,
                            const float* __restrict__ scores,
                            float* __restrict__ key, int* __restrict__ sidx,
                            int* __restrict__ pcl, float4* __restrict__ clipB) {
  int t = blockIdx.x * blockDim.x + threadIdx.x;
  if (t >= NPAD) return;
  if (t < NBOX) {
    int c = t % CLS_NUM;
    int b = t / CLS_NUM;
    int o = c * BOX_NUM + b;
    float fg = scores[2 * o];
    float s1 = scores[2 * o + 1];
    float4 bx = ((const float4*)boxes)[o];
    bool fin = isfinite(bx.x) && isfinite(bx.y) && isfinite(bx.z) &&
               isfinite(bx.w) && isfinite(fg) && isfinite(s1);
    float x1 = fminf(fmaxf(bx.x, 0.0f), IMG_Wf);
    float y1 = fminf(fmaxf(bx.y, 0.0f), IMG_Hf);
    float x2 = fminf(fmaxf(bx.z, 0.0f), IMG_Wf);
    float y2 = fminf(fmaxf(bx.w, 0.0f), IMG_Hf);
    bool valid = fin && (fg > SCORE_TH);
    key[t]  = valid ? fg : -INFINITY;
    sidx[t] = t;
    pcl[t]  = pred_cls[o];
    clipB[t] = make_float4(x1, y1, x2, y2);
  } else {
    key[t]  = -INFINITY;   // padding
    sidx[t] = t;
  }
}

// ---------------------------------------------------------------------------
// K2: one bitonic network step over (key, idx), descending order
// ---------------------------------------------------------------------------
__global__ void bitonic_kernel(float* __restrict__ key, int* __restrict__ sidx,
                               int j, int k) {
  int i   = blockIdx.x * blockDim.x + threadIdx.x;
  int ixj = i ^ j;
  if (ixj > i) {
    float ki = key[i], kj = key[ixj];
    bool descBlock = ((i & k) == 0);
    bool doSwap = descBlock ? (ki < kj) : (ki > kj);
    if (doSwap) {
      key[i] = kj; key[ixj] = ki;
      int ti = sidx[i]; sidx[i] = sidx[ixj]; sidx[ixj] = ti;
    }
  }
}

// ---------------------------------------------------------------------------
// K3: gather boxes/cls into sorted rank order; add per-class coordinate offset
// ---------------------------------------------------------------------------
__global__ void gather_kernel(const int* __restrict__ sidx,
                              const int* __restrict__ pcl,
                              const float4* __restrict__ clipB,
                              float4* __restrict__ sClip,
                              float4* __restrict__ sOff,
                              int* __restrict__ spcl) {
  int r = blockIdx.x * blockDim.x + threadIdx.x;
  if (r >= NBOX) return;
  int s = sidx[r];
  if (s < NBOX) {
    float4 b = clipB[s];
    sClip[r] = b;
    float off = (float)(s % CLS_NUM) * CLS_OFF;
    sOff[r]  = make_float4(b.x + off, b.y + off, b.z + off, b.w + off);
    spcl[r]  = pcl[s];
  } else {
    sClip[r] = make_float4(0.f, 0.f, 0.f, 0.f);
    sOff[r]  = make_float4(-1e8f, -1e8f, -1e8f, -1e8f);  // zero-area, isolated
    spcl[r]  = -1;
  }
}

// ---------------------------------------------------------------------------
// K4: suppression bitmask. One wave32 per 16x64 tile (4 WMMA subtiles).
// Pairwise area-sum (area_i + area_j) via V_WMMA_F32_16X16X4_F32:
//   A (16x4): rows m -> [area_row(m), 1, 0, 0]
//   B (4x16): cols n -> [1; area_col(n); 0; 0]   => D[m][n] = area_m + area_n
// Division-free IoU test: iou > 0.5  <=>  3*inter > asum  (asum>inter => union>0)
// Row boxes staged into LDS with global_load_async_to_lds_b128 (ASYNCcnt path).
// Ballots built per elementary compare (v_cmp -> SGPR) and combined in SALU;
// the j>i mask applies to the ~31500 tiles with c0 < r0+16 (on or below the diagonal).
// ---------------------------------------------------------------------------
__global__ void __launch_bounds__(256)
iou_mask_kernel(const float4* __restrict__ boff, unsigned* __restrict__ mask) {
  __shared__ float4 rowBox[8][16];
  __shared__ uint2  rowOut[8][16];
  int gwarp = (blockIdx.x * blockDim.x + threadIdx.x) >> 5;
  int w     = threadIdx.x >> 5;
  int lane  = threadIdx.x & 31;
  if (gwarp >= ROWT * COLT) return;   // whole-wave uniform exit
  int rt = gwarp / COLT;
  int cw = gwarp % COLT;
  int r0 = rt << 4;
  int c0 = cw << 6;
  int hi = lane >> 4;
  int lo = lane & 15;

  __builtin_prefetch(&boff[c0 + lane], 0, 0);        // global_prefetch_b8
  __builtin_prefetch(&boff[c0 + 32 + lane], 0, 0);

  // Stage the 16 row boxes into LDS via async DMA (lanes 0-15, 16B each).
  if (lane < 16) {
    unsigned lds_off = (unsigned)(size_t)(void*)&rowBox[w][lane];
    unsigned long long ga =
        (unsigned long long)(size_t)(const void*)&boff[r0 + lane];
    asm volatile("global_load_async_to_lds_b128 %0, %1, off"
                 :: "v"(lds_off), "v"(ga) : "memory");
  }
  asm volatile("s_wait_asynccnt 0x0" ::: "memory");

  // 8 row boxes per lane: rows r0 + hi*8 + v  (matches D-matrix M layout)
  float4 rb[8];
#pragma unroll
  for (int v = 0; v < 8; ++v) rb[v] = rowBox[w][hi * 8 + v];

  // A matrix: lanes 0-15 hold K=0,1 of row m=lane; lanes 16-31 hold K=2,3 (zero)
  v2f a;
  if (hi == 0) {
    float4 ra = rowBox[w][lo];
    a.x = (ra.z - ra.x) * (ra.w - ra.y);   // area of row m
    a.y = 1.0f;
  } else {
    a.x = 0.0f; a.y = 0.0f;
  }

  // 4 column subtiles: boxes + pairwise area sums via WMMA
  float4 cb[4];
  v8f asum[4];
#pragma unroll
  for (int s = 0; s < 4; ++s) {
    cb[s] = boff[c0 + s * 16 + lo];
    float areaC = (cb[s].z - cb[s].x) * (cb[s].w - cb[s].y);
    v2f b;
    if (hi == 0) { b.x = 1.0f; b.y = areaC; }
    else         { b.x = 0.0f; b.y = 0.0f;  }
    v8f c = {};
#if __has_builtin(__builtin_amdgcn_wmma_f32_16x16x4_f32)
    asum[s] = __builtin_amdgcn_wmma_f32_16x16x4_f32(
        /*neg_a=*/false, a, /*neg_b=*/false, b,
        /*c_mod=*/(short)0, c, /*reuse_a=*/false, /*reuse_b=*/false);
#else
    asum[s] = c;
#pragma unroll
    for (int v = 0; v < 8; ++v)
      asum[s][v] = (rb[v].z - rb[v].x) * (rb[v].w - rb[v].y) + areaC;
#endif
  }

  const bool diag = (c0 < r0 + 16);   // wave-uniform: tile touches or lies below the diagonal

#pragma unroll
  for (int v = 0; v < 8; ++v) {
    unsigned bal[4];
#pragma unroll
    for (int s = 0; s < 4; ++s) {
      float iw = fmaxf(fminf(rb[v].z, cb[s].z) - fmaxf(rb[v].x, cb[s].x), 0.0f);
      float ih = fmaxf(fminf(rb[v].w, cb[s].w) - fmaxf(rb[v].y, cb[s].y), 0.0f);
      float inter = iw * ih;
      float as = asum[s][v];                   // area_i + area_j
      // iou > 0.5  <=>  3*inter > asum ; (asum > inter) covers union > 0
      bal[s] = ballot32(3.0f * inter > as) & ballot32(as > inter);
    }
    if (diag) {                                // mask out j <= i
      int i = r0 + hi * 8 + v;
#pragma unroll
      for (int s = 0; s < 4; ++s)
        bal[s] &= ballot32((c0 + s * 16 + lo) > i);
    }
    // assemble the two 32-bit mask words for rows M=v and M=v+8
    unsigned w0lo = (bal[0] & 0xFFFFu) | (bal[1] << 16);
    unsigned w1lo = (bal[2] & 0xFFFFu) | (bal[3] << 16);
    unsigned w0hi = (bal[0] >> 16)     | (bal[1] & 0xFFFF0000u);
    unsigned w1hi = (bal[2] >> 16)     | (bal[3] & 0xFFFF0000u);
    if (lane == 0) {
      rowOut[w][v]     = make_uint2(w0lo, w1lo);
      rowOut[w][v + 8] = make_uint2(w0hi, w1hi);
    }
  }
  if (lane < 16) {
    uint2 m2 = rowOut[w][lane];
    ((uint2*)mask)[(size_t)(r0 + lane) * COLT + cw] = m2;  // global_store_b64
  }
}

// ---------------------------------------------------------------------------
// K5: serial greedy scan with 250-word removal bitset in LDS
// ---------------------------------------------------------------------------
__global__ void nms_scan_kernel(const unsigned* __restrict__ mask,
                                const float* __restrict__ key,
                                unsigned* __restrict__ keep) {
  __shared__ unsigned remv[COLW];
  __shared__ int aliveFlag;
  int t = threadIdx.x;
  if (t < COLW) remv[t] = 0u;
  __syncthreads();
  for (int i = 0; i < NBOX; ++i) {
    if (t == 0) {
      bool removed = (remv[i >> 5] >> (i & 31)) & 1u;
      bool alive = !removed && (key[i] > NEG_BIG);
      aliveFlag = alive ? 1 : 0;
      keep[i] = alive ? 1u : 0u;
    }
    __syncthreads();
    if (aliveFlag && t < COLW)
      remv[t] |= mask[(size_t)i * COLW + t];
    __syncthreads();
  }
}

// ---------------------------------------------------------------------------
// K6: first TOPK kept (already score-descending) -> flat output
// layout: boxes[100*4] | scores[100] | cls[100] | valid[100]
// ---------------------------------------------------------------------------
__global__ void finalize_kernel(const float* __restrict__ key,
                                const float4* __restrict__ sClip,
                                const int* __restrict__ spcl,
                                const unsigned* __restrict__ keep,
                                float* __restrict__ out) {
  if (threadIdx.x != 0 || blockIdx.x != 0) return;
  int cnt = 0;
  for (int r = 0; r < NBOX && cnt < TOPK; ++r) {
    if (keep[r]) {
      float4 b = sClip[r];
      out[cnt * 4 + 0] = b.x;
      out[cnt * 4 + 1] = b.y;
      out[cnt * 4 + 2] = b.z;
      out[cnt * 4 + 3] = b.w;
      out[4 * TOPK + cnt] = key[r];
      out[5 * TOPK + cnt] = (float)spcl[r];
      out[6 * TOPK + cnt] = 1.0f;
      ++cnt;
    }
  }
  for (; cnt < TOPK; ++cnt) {
    out[cnt * 4 + 0] = 0.0f; out[cnt * 4 + 1] = 0.0f;
    out[cnt * 4 + 2] = 0.0f; out[cnt * 4 + 3] = 0.0f;
    out[4 * TOPK + cnt] = 0.0f;
    out[5 * TOPK + cnt] = -1.0f;
    out[6 * TOPK + cnt] = 0.0f;
  }
}

// ---------------------------------------------------------------------------
extern "C" void kernel_launch(void* const* d_in, const int* in_sizes, int n_in,
                              void* d_out, int out_size, void* d_ws, size_t ws_size,
                              hipStream_t stream) {
  const int*   pred_cls = (const int*)  d_in[0];
  const float* boxes    = (const float*)d_in[1];
  const float* scores   = (const float*)d_in[2];

  char* ws = (char*)d_ws;
  float*    key   = (float*)ws;    ws += (size_t)NPAD * 4;
  int*      sidx  = (int*)ws;      ws += (size_t)NPAD * 4;
  int*      pcl   = (int*)ws;      ws += (size_t)NBOX * 4;
  float4*   clipB = (float4*)ws;   ws += (size_t)NBOX * 16;
  float4*   sClip = (float4*)ws;   ws += (size_t)NBOX * 16;
  float4*   sOff  = (float4*)ws;   ws += (size_t)NBOX * 16;
  int*      spcl  = (int*)ws;      ws += (size_t)NBOX * 4;
  unsigned* mask  = (unsigned*)ws; ws += (size_t)NBOX * COLW * 4;
  unsigned* keep  = (unsigned*)ws; ws += (size_t)NBOX * 4;

  prep_kernel<<<NPAD / 256, 256, 0, stream>>>(pred_cls, boxes, scores,
                                              key, sidx, pcl, clipB);

  for (int k = 2; k <= NPAD; k <<= 1)
    for (int j = k >> 1; j > 0; j >>= 1)
      bitonic_kernel<<<NPAD / 256, 256, 0, stream>>>(key, sidx, j, k);

  gather_kernel<<<(NBOX + 255) / 256, 256, 0, stream>>>(sidx, pcl, clipB,
                                                        sClip, sOff, spcl);

  int tiles = ROWT * COLT;                       // 62500 waves
  iou_mask_kernel<<<(tiles + 7) / 8, 256, 0, stream>>>(sOff, mask);

  nms_scan_kernel<<<1, 256, 0, stream>>>(mask, key, keep);

  finalize_kernel<<<1, 32, 0, stream>>>(key, sClip, spcl, keep, (float*)d_out);
}